// JK_NET_20469814133286
// MI455X (gfx1250) — compile-verified
//
#include <hip/hip_runtime.h>
#include <math.h>

#define N_NODES 100000
#define N_EDGES 1600000
#define DIM_H   128
#define DIM_OUT 64
#define DIM_JK  512   // 4 * DIM_H

typedef __attribute__((ext_vector_type(2))) float v2f;
typedef __attribute__((ext_vector_type(8))) float v8f;

// Fire-and-forget fp32 atomic add (native global_atomic_add_f32, no CAS loop).
__device__ __forceinline__ void atomic_add_f32(float* p, float v) {
    asm volatile("global_atomic_add_f32 %0, %1, off" :: "v"(p), "v"(v) : "memory");
}

// ---------------------------------------------------------------------------
// Degree / normalization kernels
// ---------------------------------------------------------------------------
__global__ void deg_init_kernel(float* __restrict__ deg) {
    int i = blockIdx.x * blockDim.x + threadIdx.x;
    if (i < N_NODES) deg[i] = 1.0f;   // self-loop contributes 1
}

__global__ void deg_count_kernel(const int* __restrict__ ei, float* __restrict__ deg) {
    int e = blockIdx.x * blockDim.x + threadIdx.x;
    if (e < N_EDGES) atomic_add_f32(&deg[ei[N_EDGES + e]], 1.0f);
}

__global__ void dinv_kernel(const float* __restrict__ deg, float* __restrict__ dinv) {
    int i = blockIdx.x * blockDim.x + threadIdx.x;
    if (i < N_NODES) dinv[i] = rsqrtf(deg[i]);   // deg >= 1 always
}

// ---------------------------------------------------------------------------
// WMMA fp32 GEMM:  C[nrows x NW] = A[nrows x KTOT] (row-major, lda) @ W[KTOT x NW]
// 256 threads = 8 waves; each wave owns a 16-row stripe, all NW columns.
// W staged through LDS in KC-row chunks, packed as K-PAIRS per column so each
// B fragment is a single aligned ds_load_b64:   Ws[kp][n][2] = W[2kp+j][n].
// A fragments for the whole chunk are prefetched into registers up front.
// Uses V_WMMA_F32_16X16X4_F32 (fp32 in, fp32 accumulate).
// ---------------------------------------------------------------------------
template <int KTOT, int NW, int KC>
__global__ void wmma_gemm_kernel(const float* __restrict__ A, int lda,
                                 const float* __restrict__ W,
                                 float* __restrict__ C, int ldc,
                                 const float* __restrict__ bias, int nrows) {
    constexpr int NT = NW / 16;          // column tiles
    constexpr int KT = KC / 4;           // k-tiles per chunk
    __shared__ __align__(16) float Ws[KC * NW];   // packed: [KC/2][NW][2]

    const int tid   = threadIdx.x;
    const int wave  = tid >> 5;
    const int lane  = tid & 31;
    const int nl    = lane & 15;         // column-within-tile / row-within-tile
    const int khalf = lane >> 4;         // K half selector for A/B frags
    const int rowBase = blockIdx.x * 128 + wave * 16;
    const int mRow    = rowBase + nl;                        // A row this lane loads
    const int mClamp  = (mRow < nrows) ? mRow : (nrows - 1); // padded rows never stored
    const float* Arow = A + (size_t)mClamp * lda;

    v8f zero = {};
    v8f acc[NT];
#pragma unroll
    for (int t = 0; t < NT; ++t) acc[t] = zero;

    for (int c = 0; c < KTOT / KC; ++c) {
        // stage W chunk into LDS, interleaving K-pairs per column
        for (int i = tid; i < KC * NW; i += 256) {
            int k = i / NW, n = i - k * NW;
            Ws[(k >> 1) * (NW * 2) + n * 2 + (k & 1)] = W[(size_t)c * KC * NW + i];
        }
        __syncthreads();

        // prefetch all A fragments for this chunk (pipelined global_load_b64)
        v2f afrag[KT];
#pragma unroll
        for (int kt = 0; kt < KT; ++kt)
            afrag[kt] = *(const v2f*)(Arow + c * KC + kt * 4 + khalf * 2);

#pragma unroll
        for (int kt = 0; kt < KT; ++kt) {
            const int kp = kt * 2 + khalf;               // packed K-pair row
            const float* bbase = &Ws[kp * (NW * 2)];
#pragma unroll
            for (int nt = 0; nt < NT; ++nt) {
                v2f b = *(const v2f*)(bbase + (nt * 16 + nl) * 2);  // one ds_load_b64
                acc[nt] = __builtin_amdgcn_wmma_f32_16x16x4_f32(
                    false, afrag[kt], false, b, (short)0, acc[nt], false, false);
            }
        }
        __syncthreads();
    }

    // C/D layout: VGPR v, lanes 0-15 -> M=v ; lanes 16-31 -> M=v+8 ; N = nl
#pragma unroll
    for (int nt = 0; nt < NT; ++nt) {
        const int col = nt * 16 + nl;
        const float bv = bias ? bias[col] : 0.0f;
#pragma unroll
        for (int v = 0; v < 8; ++v) {
            const int row = rowBase + v + 8 * khalf;
            if (row < nrows)
                C[(size_t)row * ldc + col] = acc[nt][v] + bv;
        }
    }
}

// ---------------------------------------------------------------------------
// agg[i,:] = tmp[i,:] * dinv[i]^2      (self-loop edge, weight dinv*dinv)
// ---------------------------------------------------------------------------
__global__ void self_loop_init_kernel(const float* __restrict__ tmp,
                                      const float* __restrict__ dinv,
                                      float* __restrict__ agg) {
    int gid = blockIdx.x * blockDim.x + threadIdx.x;
    if (gid >= N_NODES * 32) return;
    int node = gid >> 5;
    int c    = (gid & 31) * 4;
    float w = dinv[node];
    w *= w;
    float4 v = *(const float4*)(tmp + (size_t)node * DIM_H + c);
    v.x *= w; v.y *= w; v.z *= w; v.w *= w;
    *(float4*)(agg + (size_t)node * DIM_H + c) = v;
}

// ---------------------------------------------------------------------------
// Edge scatter: one wave per edge. Lane loads float4 of src row (512B/wave
// coalesced), scales by dinv[src]*dinv[dst], atomic-adds into dst row with
// native global_atomic_add_f32. tmp/agg (51.2MB each) are L2-resident.
// ---------------------------------------------------------------------------
__global__ void edge_scatter_kernel(const int* __restrict__ ei,
                                    const float* __restrict__ dinv,
                                    const float* __restrict__ tmp,
                                    float* __restrict__ agg) {
    int gid  = blockIdx.x * blockDim.x + threadIdx.x;
    int e    = gid >> 5;
    int lane = gid & 31;
    if (e >= N_EDGES) return;
    int s = ei[e];
    int d = ei[N_EDGES + e];
    float w = dinv[s] * dinv[d];
    float4 v = *(const float4*)(tmp + (size_t)s * DIM_H + lane * 4);
    float* p = agg + (size_t)d * DIM_H + lane * 4;
    asm volatile("global_atomic_add_f32 %0, %1, off"          :: "v"(p), "v"(v.x * w) : "memory");
    asm volatile("global_atomic_add_f32 %0, %1, off offset:4" :: "v"(p), "v"(v.y * w) : "memory");
    asm volatile("global_atomic_add_f32 %0, %1, off offset:8" :: "v"(p), "v"(v.z * w) : "memory");
    asm volatile("global_atomic_add_f32 %0, %1, off offset:12":: "v"(p), "v"(v.w * w) : "memory");
}

// ---------------------------------------------------------------------------
// jk[i, col_off + c] = relu(agg[i,c] + bias[c])  — writes this layer's slice
// of the JK concat buffer; that slice IS the next layer's input (ld = 512).
// ---------------------------------------------------------------------------
__global__ void finalize_relu_kernel(const float* __restrict__ agg,
                                     const float* __restrict__ bias,
                                     float* __restrict__ jk, int col_off) {
    int gid = blockIdx.x * blockDim.x + threadIdx.x;
    if (gid >= N_NODES * 32) return;
    int node = gid >> 5;
    int c    = (gid & 31) * 4;
    float4 v = *(const float4*)(agg + (size_t)node * DIM_H + c);
    float4 b = *(const float4*)(bias + c);
    v.x = fmaxf(v.x + b.x, 0.0f);
    v.y = fmaxf(v.y + b.y, 0.0f);
    v.z = fmaxf(v.z + b.z, 0.0f);
    v.w = fmaxf(v.w + b.w, 0.0f);
    *(float4*)(jk + (size_t)node * DIM_JK + col_off + c) = v;
}

// ---------------------------------------------------------------------------
// In-place log-softmax over rows of 64. One wave32 per row, 2 elems/lane,
// shfl_xor reductions (wave32 widths).
// ---------------------------------------------------------------------------
__global__ void log_softmax_kernel(float* __restrict__ logits) {
    int gid  = blockIdx.x * blockDim.x + threadIdx.x;
    int row  = gid >> 5;
    int lane = gid & 31;
    if (row >= N_NODES) return;
    float* p = logits + (size_t)row * DIM_OUT;
    float v0 = p[lane];
    float v1 = p[lane + 32];
    float m = fmaxf(v0, v1);
#pragma unroll
    for (int off = 16; off > 0; off >>= 1) m = fmaxf(m, __shfl_xor(m, off, 32));
    float s = expf(v0 - m) + expf(v1 - m);
#pragma unroll
    for (int off = 16; off > 0; off >>= 1) s += __shfl_xor(s, off, 32);
    float lse = m + logf(s);
    p[lane]      = v0 - lse;
    p[lane + 32] = v1 - lse;
}

// ---------------------------------------------------------------------------
// Host-side launch. Inputs (setup_inputs order):
//  0:x [N,128] f32   1:edge_index [2,E] int   2..9: W0,b0,..,W3,b3
//  10:lin_W [512,64] 11:lin_b [64]            out: [N,64] f32 log-probs
// ---------------------------------------------------------------------------
extern "C" void kernel_launch(void* const* d_in, const int* in_sizes, int n_in,
                              void* d_out, int out_size, void* d_ws, size_t ws_size,
                              hipStream_t stream) {
    const float* x    = (const float*)d_in[0];
    const int*   ei   = (const int*)d_in[1];
    const float* W[4] = {(const float*)d_in[2], (const float*)d_in[4],
                         (const float*)d_in[6], (const float*)d_in[8]};
    const float* b[4] = {(const float*)d_in[3], (const float*)d_in[5],
                         (const float*)d_in[7], (const float*)d_in[9]};
    const float* linW = (const float*)d_in[10];
    const float* linb = (const float*)d_in[11];
    float* out = (float*)d_out;

    float* ws   = (float*)d_ws;
    float* deg  = ws;                               // [N]
    float* dinv = deg + N_NODES;                    // [N]
    float* tmp  = dinv + N_NODES;                   // [N,128]  h @ W
    float* agg  = tmp + (size_t)N_NODES * DIM_H;    // [N,128]  scatter accum
    float* jk   = agg + (size_t)N_NODES * DIM_H;    // [N,512]  JK concat (layer slices)

    const int T = 256;
    const int nodeBlk  = (N_NODES + T - 1) / T;
    const int edgeBlk  = (N_EDGES + T - 1) / T;
    const int nodeWBlk = (N_NODES * 32 + T - 1) / T;   // one wave per node
    const int edgeWBlk = (int)(((size_t)N_EDGES * 32 + T - 1) / T);
    const int gemmBlk  = (N_NODES + 127) / 128;        // 128 rows per block

    // normalization: deg = 1 + in-degree ; dinv = rsqrt(deg)
    deg_init_kernel<<<nodeBlk, T, 0, stream>>>(deg);
    deg_count_kernel<<<edgeBlk, T, 0, stream>>>(ei, deg);
    dinv_kernel<<<nodeBlk, T, 0, stream>>>(deg, dinv);

    // 4 GCN layers; layer l writes jk[:, l*128:(l+1)*128]
    for (int l = 0; l < 4; ++l) {
        const float* hin = (l == 0) ? x : (jk + (size_t)(l - 1) * DIM_H);
        const int    lda = (l == 0) ? DIM_H : DIM_JK;
        wmma_gemm_kernel<128, 128, 64><<<gemmBlk, T, 0, stream>>>(
            hin, lda, W[l], tmp, DIM_H, nullptr, N_NODES);
        self_loop_init_kernel<<<nodeWBlk, T, 0, stream>>>(tmp, dinv, agg);
        edge_scatter_kernel<<<edgeWBlk, T, 0, stream>>>(ei, dinv, tmp, agg);
        finalize_relu_kernel<<<nodeWBlk, T, 0, stream>>>(agg, b[l], jk, l * DIM_H);
    }

    // logits = jk @ lin_W + lin_b ; then in-place log-softmax
    wmma_gemm_kernel<512, 64, 128><<<gemmBlk, T, 0, stream>>>(
        jk, DIM_JK, linW, out, DIM_OUT, linb, N_NODES);
    log_softmax_kernel<<<nodeWBlk, T, 0, stream>>>(out);
}